// SimGNN_student_57784490000393
// MI455X (gfx1250) — compile-verified
//
#include <hip/hip_runtime.h>

typedef __attribute__((ext_vector_type(2))) float v2f;
typedef __attribute__((ext_vector_type(8))) float v8f;

// ---------------------------------------------------------------------------
// Small utility kernels
// ---------------------------------------------------------------------------
__global__ void fill_kernel(float* __restrict__ p, float v, long n) {
    long t = (long)blockIdx.x * blockDim.x + threadIdx.x;
    if (t < n) p[t] = v;
}

__global__ void deg_kernel(const int* __restrict__ dst, float* __restrict__ deg, int E) {
    int t = blockIdx.x * blockDim.x + threadIdx.x;
    if (t < E) atomicAdd(&deg[dst[t]], 1.0f);
}

__global__ void rsqrt_kernel(float* __restrict__ d, int n) {
    int t = blockIdx.x * blockDim.x + threadIdx.x;
    if (t < n) d[t] = rsqrtf(d[t]);
}

// W[64 x F] row-major  ->  Wt[F x 64] row-major (transpose, stays fp32)
__global__ void wt_kernel(const float* __restrict__ W, float* __restrict__ Wt, int F) {
    int t = blockIdx.x * blockDim.x + threadIdx.x;
    if (t < 64 * F) {
        int r = t / F, c = t % F;
        Wt[c * 64 + r] = W[t];
    }
}

// ---------------------------------------------------------------------------
// fp32 WMMA GEMM, fused self-loop:
//   H[n x F]   = X[n x 64] @ W[64 x F]
//   OUT[n x F] = H * dinv^2            (GCN self-loop contribution)
//
// Full fp32 via v_wmma_f32_16x16x4_f32 (K stepped by 4, 16 steps).
// X tile (64x64) and Wt tile (Fx64) are staged into LDS with
// global_load_async_to_lds_b128 (ASYNCcnt) -- no VGPR staging traffic.
// Fragment loads are aligned float2 from LDS: lane m reads banks {4m,4m+1}
// (row stride 68 dwords) -> conflict-free.  One wave per 16-row x F strip (F/16 C tiles),
// 128-thread block = 64 rows x F cols.
// ---------------------------------------------------------------------------
template <int F>
__global__ void gemm_wmma_kernel(const float* __restrict__ X,
                                 const float* __restrict__ Wt,
                                 const float* __restrict__ dinv,
                                 float* __restrict__ H,
                                 float* __restrict__ OUT, int n) {
    __shared__ __align__(16) float sX[64][68];
    __shared__ __align__(16) float sW[F][68];

    const int tid  = threadIdx.x;        // 128 threads = 4 waves (wave32)
    const int lane = tid & 31;
    const int wave = tid >> 5;
    const int row0 = blockIdx.x * 64;
    const int m    = lane & 15;
    const int ksel = (lane >= 16) ? 2 : 0;   // A/B VGPR layout: K0/K2, K1/K3

    // ---- async-stage X tile (64 rows x 64 k, fp32) ----
#pragma unroll
    for (int i = 0; i < 8; ++i) {
        int e = tid + i * 128;
        int r = e >> 4, c = (e & 15) << 2;
        int gr = row0 + r;
        if (gr >= n) gr = n - 1;         // clamp: garbage rows never stored
        const float* gp = X + (long)gr * 64 + c;
        unsigned lo = (unsigned)(size_t)&sX[r][c];
        asm volatile("global_load_async_to_lds_b128 %0, %1, off"
                     :: "v"(lo), "v"(gp) : "memory");
    }
    // ---- async-stage Wt tile (F rows x 64 k, fp32) ----
#pragma unroll
    for (int i = 0; i < F / 8; ++i) {
        int e = tid + i * 128;
        int r = e >> 4, c = (e & 15) << 2;
        const float* gp = Wt + r * 64 + c;
        unsigned lo = (unsigned)(size_t)&sW[r][c];
        asm volatile("global_load_async_to_lds_b128 %0, %1, off"
                     :: "v"(lo), "v"(gp) : "memory");
    }
    asm volatile("s_wait_asynccnt 0" ::: "memory");
    __syncthreads();

    constexpr int NT = F >> 4;
    v8f acc[NT];
#pragma unroll
    for (int i = 0; i < NT; ++i) acc[i] = (v8f){};

    const float* ar = &sX[wave * 16 + m][ksel];
#pragma unroll
    for (int kt = 0; kt < 16; ++kt) {
        const int kk = kt * 4;
        v2f a;
        a.x = ar[kk];
        a.y = ar[kk + 1];
#pragma unroll
        for (int nt = 0; nt < NT; ++nt) {
            const float* br = &sW[nt * 16 + m][kk + ksel];
            v2f b;
            b.x = br[0];
            b.y = br[1];
            acc[nt] = __builtin_amdgcn_wmma_f32_16x16x4_f32(
                false, a, false, b, (short)0, acc[nt], false, false);
        }
    }

    // C/D layout: lane l, vgpr v -> (M = v + (l>=16 ? 8:0), N = l&15)
    const int hi8   = (lane >= 16) ? 8 : 0;
    const int rbase = row0 + wave * 16 + hi8;
    float dv[8];
#pragma unroll
    for (int v = 0; v < 8; ++v) {
        int row = rbase + v;
        dv[v] = (row < n) ? dinv[row] : 0.0f;
    }
#pragma unroll
    for (int nt = 0; nt < NT; ++nt) {
        const int col = nt * 16 + m;
#pragma unroll
        for (int v = 0; v < 8; ++v) {
            int row = rbase + v;
            if (row < n) {
                float val = acc[nt][v];
                H[(long)row * F + col]   = val;
                OUT[(long)row * F + col] = val * dv[v] * dv[v];
            }
        }
    }
}

// ---------------------------------------------------------------------------
// GCN edge aggregation: OUT[dst] += H[src] * dinv[src]*dinv[dst]
// float4 per thread, global_atomic_add_f32; bandwidth-bound hot loop.
// ---------------------------------------------------------------------------
template <int F>
__global__ void edge_agg_kernel(const float* __restrict__ H,
                                const int* __restrict__ src,
                                const int* __restrict__ dst,
                                const float* __restrict__ dinv,
                                float* __restrict__ OUT, int E) {
    constexpr int CH = F / 4;            // float4 chunks per edge
    long t = (long)blockIdx.x * blockDim.x + threadIdx.x;
    if (t >= (long)E * CH) return;
    int e = (int)(t / CH);
    int c = (int)(t % CH) * 4;
    int s = src[e], d = dst[e];
    float nrm = dinv[s] * dinv[d];
    const float4 hv = *(const float4*)(H + (long)s * F + c);
    float* o = OUT + (long)d * F + c;
    atomicAdd(o + 0, hv.x * nrm);
    atomicAdd(o + 1, hv.y * nrm);
    atomicAdd(o + 2, hv.z * nrm);
    atomicAdd(o + 3, hv.w * nrm);
}

template <int F, bool RELU>
__global__ void bias_act_kernel(float* __restrict__ O, const float* __restrict__ b, int n) {
    long t = (long)blockIdx.x * blockDim.x + threadIdx.x;
    if (t >= (long)n * F) return;
    float v = O[t] + b[t % F];
    O[t] = RELU ? fmaxf(v, 0.0f) : v;
}

// ---------------------------------------------------------------------------
// Attention pooling.  mean(x@Watt) == (colsum(x)/N)@Watt, so: colsum -> tiny
// matvec+tanh -> one coalesced pass computing sigmoid(x.gc) and x^T s.
// ---------------------------------------------------------------------------
__global__ void colsum_kernel(const float* __restrict__ X, float* __restrict__ cs, int n) {
    __shared__ float sbuf[256];
    const int col = threadIdx.x & 31;
    const int rowoff = threadIdx.x >> 5;
    float acc = 0.0f;
    for (long i = (long)blockIdx.x * 8 + rowoff; i < n; i += (long)gridDim.x * 8)
        acc += X[i * 32 + col];
    sbuf[threadIdx.x] = acc;
    __syncthreads();
    if (threadIdx.x < 32) {
        float s = 0.0f;
        for (int w = 0; w < 8; ++w) s += sbuf[w * 32 + col];
        atomicAdd(&cs[col], s);
    }
}

__global__ void gc_kernel(const float* __restrict__ cs, const float* __restrict__ Watt,
                          float* __restrict__ gc, float invN) {
    int j = threadIdx.x;  // 32 threads
    float s = 0.0f;
    for (int i = 0; i < 32; ++i) s += cs[i] * invN * Watt[i * 32 + j];
    gc[j] = tanhf(s);
}

__global__ void pool_kernel(const float* __restrict__ X, const float* __restrict__ gc,
                            float* __restrict__ p, int n) {
    const int lane = threadIdx.x & 31;
    const int wid  = (blockIdx.x * blockDim.x + threadIdx.x) >> 5;
    const int nw   = (gridDim.x * blockDim.x) >> 5;
    const float g  = gc[lane];
    float acc = 0.0f;
    for (int i = wid; i < n; i += nw) {
        float v = X[(long)i * 32 + lane];
        float d = v * g;
#pragma unroll
        for (int off = 16; off > 0; off >>= 1) d += __shfl_xor(d, off, 32);
        float s = 1.0f / (1.0f + __expf(-d));
        acc += v * s;
    }
    atomicAdd(&p[lane], acc);
}

// ---------------------------------------------------------------------------
// NTN + MLP head: bilinear + block + MLP + sigmoid.  p1/p2 live in d_out.
// ---------------------------------------------------------------------------
__global__ void head_kernel(const float* __restrict__ Wtnn, const float* __restrict__ Wblock,
                            const float* __restrict__ btnn, const float* __restrict__ Wfc,
                            const float* __restrict__ bfc, const float* __restrict__ Wsc,
                            const float* __restrict__ bsc, float* __restrict__ out) {
    __shared__ float scores[16];
    __shared__ float hbuf[16];
    const float* p1 = out + 1;
    const float* p2 = out + 33;
    const int t = threadIdx.x;  // launched with 32
    if (t < 16) {
        float bil = 0.0f;
        for (int i = 0; i < 32; ++i) {
            float a = p1[i];
            const float* w = Wtnn + (long)i * 32 * 16 + t;
            for (int j = 0; j < 32; ++j) bil += a * w[j * 16] * p2[j];
        }
        float blk = 0.0f;
        for (int q = 0; q < 32; ++q)
            blk += Wblock[t * 64 + q] * p1[q] + Wblock[t * 64 + 32 + q] * p2[q];
        float sc = bil + blk + btnn[t];
        scores[t] = sc > 0.0f ? sc : 0.0f;
    }
    __syncthreads();
    if (t < 16) {
        float hv = bfc[t];
        for (int k = 0; k < 16; ++k) hv += scores[k] * Wfc[k * 16 + t];
        hbuf[t] = hv > 0.0f ? hv : 0.0f;
    }
    __syncthreads();
    if (t == 0) {
        float sv = bsc[0];
        for (int b = 0; b < 16; ++b) sv += hbuf[b] * Wsc[b];
        out[0] = 1.0f / (1.0f + __expf(-sv));
    }
}

// ---------------------------------------------------------------------------
// Host-side orchestration
// ---------------------------------------------------------------------------
static inline int cdiv(long a, long b) { return (int)((a + b - 1) / b); }

extern "C" void kernel_launch(void* const* d_in, const int* in_sizes, int n_in,
                              void* d_out, int out_size, void* d_ws, size_t ws_size,
                              hipStream_t stream) {
    const int* ei[2]     = {(const int*)d_in[0], (const int*)d_in[2]};
    const float* feat[2] = {(const float*)d_in[1], (const float*)d_in[3]};
    const float* W1 = (const float*)d_in[4];
    const float* b1 = (const float*)d_in[5];
    const float* W2 = (const float*)d_in[6];
    const float* b2 = (const float*)d_in[7];
    const float* W3 = (const float*)d_in[8];
    const float* b3 = (const float*)d_in[9];
    const float* Watt   = (const float*)d_in[10];
    const float* Wtnn   = (const float*)d_in[11];
    const float* Wblock = (const float*)d_in[12];
    const float* btnn   = (const float*)d_in[13];
    const float* Wfc    = (const float*)d_in[14];
    const float* bfc    = (const float*)d_in[15];
    const float* Wsc    = (const float*)d_in[16];
    const float* bsc    = (const float*)d_in[17];

    const int N = in_sizes[1] / 64;   // 100000
    const int E = in_sizes[0] / 2;    // 1200000
    float* out = (float*)d_out;

    // workspace layout (floats)
    float* ws   = (float*)d_ws;
    float* dinv = ws;                                  // N
    float* buf1 = ws + N;                              // N*64 (GEMM output h)
    float* buf2 = buf1 + (size_t)N * 64;               // N*64
    float* buf3 = buf2 + (size_t)N * 64;               // N*64
    float* cs   = buf3 + (size_t)N * 64;               // 32
    float* gc   = cs + 32;                             // 32
    float* wt   = gc + 32;                             // 4096 (transposed W, fp32)

    // zero p1/p2 accumulators in d_out
    fill_kernel<<<1, 64, 0, stream>>>(out + 1, 0.0f, 64);

    for (int g = 0; g < 2; ++g) {
        const int* src = ei[g];
        const int* dst = ei[g] + E;

        // degree (self-loop counted) -> dinv = deg^{-1/2}
        fill_kernel<<<cdiv(N, 256), 256, 0, stream>>>(dinv, 1.0f, N);
        deg_kernel<<<cdiv(E, 256), 256, 0, stream>>>(dst, dinv, E);
        rsqrt_kernel<<<cdiv(N, 256), 256, 0, stream>>>(dinv, N);

        // ---- layer 1: X=feat, F=64, out -> buf2 (relu) ----
        wt_kernel<<<cdiv(64 * 64, 256), 256, 0, stream>>>(W1, wt, 64);
        gemm_wmma_kernel<64><<<cdiv(N, 64), 128, 0, stream>>>(feat[g], wt, dinv, buf1, buf2, N);
        edge_agg_kernel<64><<<cdiv((long)E * 16, 256), 256, 0, stream>>>(buf1, src, dst, dinv, buf2, E);
        bias_act_kernel<64, true><<<cdiv((long)N * 64, 256), 256, 0, stream>>>(buf2, b1, N);

        // ---- layer 2: X=buf2, F=64, out -> buf3 (relu) ----
        wt_kernel<<<cdiv(64 * 64, 256), 256, 0, stream>>>(W2, wt, 64);
        gemm_wmma_kernel<64><<<cdiv(N, 64), 128, 0, stream>>>(buf2, wt, dinv, buf1, buf3, N);
        edge_agg_kernel<64><<<cdiv((long)E * 16, 256), 256, 0, stream>>>(buf1, src, dst, dinv, buf3, E);
        bias_act_kernel<64, true><<<cdiv((long)N * 64, 256), 256, 0, stream>>>(buf3, b2, N);

        // ---- layer 3: X=buf3, F=32, out -> buf2 (no relu) ----
        wt_kernel<<<cdiv(64 * 32, 256), 256, 0, stream>>>(W3, wt, 32);
        gemm_wmma_kernel<32><<<cdiv(N, 64), 128, 0, stream>>>(buf3, wt, dinv, buf1, buf2, N);
        edge_agg_kernel<32><<<cdiv((long)E * 8, 256), 256, 0, stream>>>(buf1, src, dst, dinv, buf2, E);
        bias_act_kernel<32, false><<<cdiv((long)N * 32, 256), 256, 0, stream>>>(buf2, b3, N);

        // ---- attention pooling: p_g = x^T sigmoid(x . tanh(mean(x@Watt))) ----
        fill_kernel<<<1, 32, 0, stream>>>(cs, 0.0f, 32);
        colsum_kernel<<<512, 256, 0, stream>>>(buf2, cs, N);
        gc_kernel<<<1, 32, 0, stream>>>(cs, Watt, gc, 1.0f / (float)N);
        pool_kernel<<<256, 256, 0, stream>>>(buf2, gc, out + 1 + 32 * g, N);
    }

    // ---- NTN + MLP head (reads p1/p2 from d_out, writes score to d_out[0]) ----
    head_kernel<<<1, 32, 0, stream>>>(Wtnn, Wblock, btnn, Wfc, bfc, Wsc, bsc, out);
}